// MultiFieldEncoder_39298950758413
// MI455X (gfx1250) — compile-verified
//
#include <hip/hip_runtime.h>
#include <math.h>

// ---------------------------------------------------------------------------
// MultiFieldEncoder for MI455X (gfx1250, wave32, WMMA).
// All GEMMs use V_WMMA_F32_16X16X4_F32 (full fp32 precision, D = A(16x4)*B(4x16)+C).
// ---------------------------------------------------------------------------

typedef float v2f __attribute__((ext_vector_type(2)));
typedef float v8f __attribute__((ext_vector_type(8)));

#define DD      128
#define BB      2
#define NNODES  50000
#define MEDGES  800000

__device__ __forceinline__ v8f wmma_f32_k4(v2f a, v2f b, v8f c) {
  // 8-arg form: (neg_a, A, neg_b, B, c_mod, C, reuse_a, reuse_b)
  return __builtin_amdgcn_wmma_f32_16x16x4_f32(false, a, false, b, (short)0, c,
                                               false, false);
}

// silu via hardware reciprocal (v_rcp_f32, ~1 ulp) instead of IEEE division:
// mul, v_exp_f32, add, v_rcp_f32, mul  (~5 VALU ops vs ~12 for exact div).
__device__ __forceinline__ float silu_f(float x) {
  return x * __builtin_amdgcn_rcpf(1.0f + __expf(-x));
}

// Pre-swizzle a [rows<=128][128] weight matrix into WMMA-B fragment order in LDS.
// Fragment (ks = r>>2, nn = c>>4) holds a 4x16 block; each lane's two dwords are
// contiguous so a B-operand is one aligned 8-byte ds read, bank-conflict free.
// B layout: lanes 0-15 -> rows (4ks+0,4ks+1), lanes 16-31 -> rows (4ks+2,4ks+3),
// N = lane&15 (mirrors the documented 32-bit A layout).
__device__ __forceinline__ void stage_weight_frags(float* lds,
                                                   const float* __restrict__ W,
                                                   int nelem, int tid, int nthreads) {
  for (int i = tid; i < nelem; i += nthreads) {
    const int r = i >> 7, c = i & 127;
    const int ks = r >> 2, nn = c >> 4, kk = r & 3, col = c & 15;
    const int lane = ((kk & 2) << 3) | col;
    lds[((ks * 8 + nn) << 6) + (lane << 1) + (kk & 1)] = W[i];
  }
}

// acc[nn] += H(16x128, held as 8 C-layout tiles) @ W(128x128, LDS fragment order).
// st = per-wave 16x17 staging buffer used to convert C-layout -> A-fragments.
__device__ __forceinline__ void gemm_h_times_w(float* st, const float* ldsW,
                                               int lane, const v8f h[8], v8f acc[8]) {
  const int hf = lane >> 4;
  const int col = lane & 15;
  const int r = col;
  for (int kt = 0; kt < 8; ++kt) {
#pragma unroll
    for (int v = 0; v < 8; ++v) st[(v + 8 * hf) * 17 + col] = h[kt][v];
    v2f af[4];
#pragma unroll
    for (int k4 = 0; k4 < 4; ++k4) {
      af[k4].x = st[r * 17 + 4 * k4 + 2 * hf];
      af[k4].y = st[r * 17 + 4 * k4 + 2 * hf + 1];
    }
#pragma unroll
    for (int k4 = 0; k4 < 4; ++k4) {
      const int ks = kt * 4 + k4;
#pragma unroll
      for (int nn = 0; nn < 8; ++nn) {
        const v2f bf = *(const v2f*)(ldsW + ((ks * 8 + nn) << 6) + (lane << 1));
        acc[nn] = wmma_f32_k4(af[k4], bf, acc[nn]);
      }
    }
  }
}

// ---------------------------------------------------------------------------
// Kernel 1: time/cond encoder MLPs -> bias[2][128] in d_ws (tiny, VALU only).
// ---------------------------------------------------------------------------
__global__ void __launch_bounds__(128) bias_kernel(
    const float* __restrict__ time_i, const float* __restrict__ cond,
    const float* __restrict__ tw1, const float* __restrict__ tb1,
    const float* __restrict__ tw2, const float* __restrict__ tb2,
    const float* __restrict__ cw1, const float* __restrict__ cb1,
    const float* __restrict__ cw2, const float* __restrict__ cb2,
    float* __restrict__ biasOut) {
  __shared__ float h[DD];
  const int b = blockIdx.x, t = threadIdx.x;

  float acc = tb1[t];
  for (int k = 0; k < 11; ++k) acc += time_i[b * 11 + k] * tw1[k * DD + t];
  h[t] = silu_f(acc);
  __syncthreads();
  float o = tb2[t];
  for (int k = 0; k < DD; ++k) o += h[k] * tw2[k * DD + t];
  __syncthreads();

  acc = cb1[t];
  for (int k = 0; k < 12; ++k) acc += cond[b * 12 + k] * cw1[k * DD + t];
  h[t] = silu_f(acc);
  __syncthreads();
  float o2 = cb2[t];
  for (int k = 0; k < DD; ++k) o2 += h[k] * cw2[k * DD + t];

  biasOut[b * DD + t] = o + o2;
}

// ---------------------------------------------------------------------------
// Kernel 2: per-field node MLP  V = silu(x@w1+b1)@w2 + b2 + biasTC[b]
// One wave per 16-node tile. Layer-1 is a single 16x16x4 WMMA per N-tile.
// ---------------------------------------------------------------------------
__global__ void __launch_bounds__(256) node_field_kernel(
    const float* __restrict__ node_pos, const float* __restrict__ state_in,
    const float* __restrict__ w1, const float* __restrict__ b1,
    const float* __restrict__ w2, const float* __restrict__ b2,
    const float* __restrict__ biasTC, float* __restrict__ outV, int field) {
  __shared__ __align__(16) float ldsW2[DD * DD];
  __shared__ __align__(16) float stageAll[8][16 * 17];
  const int tid = threadIdx.x;
  const int wave = tid >> 5, lane = tid & 31;
  const int hf = lane >> 4, col = lane & 15;

  stage_weight_frags(ldsW2, w2, DD * DD, tid, 256);
  __syncthreads();

  v2f w1f[8];
  float b1r[8], b2r[8], btc0[8], btc1[8];
#pragma unroll
  for (int nn = 0; nn < 8; ++nn) {
    w1f[nn].x = w1[(2 * hf) * DD + nn * 16 + col];
    w1f[nn].y = w1[(2 * hf + 1) * DD + nn * 16 + col];
    b1r[nn] = b1[nn * 16 + col];
    b2r[nn] = b2[nn * 16 + col];
    btc0[nn] = biasTC[nn * 16 + col];
    btc1[nn] = biasTC[DD + nn * 16 + col];
  }

  const int tilesPerB = NNODES / 16;
  const int nTiles = BB * tilesPerB;
  float* st = stageAll[wave];

  for (int tile = blockIdx.x * 8 + wave; tile < nTiles; tile += gridDim.x * 8) {
    const int b = tile / tilesPerB;
    const int n0 = (tile - b * tilesPerB) * 16;
    const long nb = (long)b * NNODES + n0 + (lane & 15);

    // A operand (16x4 features) built directly in A-layout.
    v2f a;
    if (hf == 0) { a.x = state_in[nb * 2 + field]; a.y = node_pos[nb * 3 + 0]; }
    else         { a.x = node_pos[nb * 3 + 1];     a.y = node_pos[nb * 3 + 2]; }

    v8f h[8], acc[8];
#pragma unroll
    for (int nn = 0; nn < 8; ++nn) {
      v8f c = {};
      c = wmma_f32_k4(a, w1f[nn], c);
#pragma unroll
      for (int e = 0; e < 8; ++e) h[nn][e] = silu_f(c[e] + b1r[nn]);
      const float bout = b2r[nn] + ((b == 0) ? btc0[nn] : btc1[nn]);
#pragma unroll
      for (int e = 0; e < 8; ++e) acc[nn][e] = bout;
    }

    gemm_h_times_w(st, ldsW2, lane, h, acc);

    const long obase = ((long)b * NNODES + n0) * DD;
#pragma unroll
    for (int nn = 0; nn < 8; ++nn)
#pragma unroll
      for (int v = 0; v < 8; ++v)
        outV[obase + (long)(v + 8 * hf) * DD + nn * 16 + col] = acc[nn][v];
  }
}

// ---------------------------------------------------------------------------
// Kernel 3: gated cross-field exchange (in-place on V0/V1 output regions).
// Vx0 = V0 + tanh(g0)*silu(V1@ex0+b0); Vx1 = V1 + tanh(g1)*silu(V0@ex1+b1)
// Both 128x128 weights live in LDS (128KB) in fragment order.
// ---------------------------------------------------------------------------
__global__ void __launch_bounds__(256) exchange_kernel(
    float* __restrict__ V0, float* __restrict__ V1,
    const float* __restrict__ ex0w, const float* __restrict__ ex0b,
    const float* __restrict__ ex1w, const float* __restrict__ ex1b,
    const float* __restrict__ gate0, const float* __restrict__ gate1) {
  __shared__ __align__(16) float ldsW0[DD * DD];
  __shared__ __align__(16) float ldsW1[DD * DD];
  __shared__ __align__(16) float stageAll[8][16 * 17];
  const int tid = threadIdx.x, wave = tid >> 5, lane = tid & 31;
  const int hf = lane >> 4, col = lane & 15, r = col;

  stage_weight_frags(ldsW0, ex0w, DD * DD, tid, 256);
  stage_weight_frags(ldsW1, ex1w, DD * DD, tid, 256);
  __syncthreads();

  const float t0 = tanhf(gate0[0]);
  const float t1 = tanhf(gate1[0]);
  float b0r[8], b1r[8];
#pragma unroll
  for (int nn = 0; nn < 8; ++nn) {
    b0r[nn] = ex0b[nn * 16 + col];
    b1r[nn] = ex1b[nn * 16 + col];
  }

  const int tilesPerB = NNODES / 16;
  const int nTiles = BB * tilesPerB;
  float* st = stageAll[wave];

  for (int tile = blockIdx.x * 8 + wave; tile < nTiles; tile += gridDim.x * 8) {
    const int b = tile / tilesPerB;
    const int n0 = (tile - b * tilesPerB) * 16;
    const long rbase = ((long)b * NNODES + n0) * DD;

    v8f acc0[8], acc1[8];
#pragma unroll
    for (int nn = 0; nn < 8; ++nn)
#pragma unroll
      for (int e = 0; e < 8; ++e) { acc0[nn][e] = b0r[nn]; acc1[nn][e] = b1r[nn]; }

    for (int kt = 0; kt < 8; ++kt) {
      // ---- acc0 += V1-tile(kt) @ ex0 ----
#pragma unroll
      for (int v = 0; v < 8; ++v)
        st[(v + 8 * hf) * 17 + col] =
            V1[rbase + (long)(v + 8 * hf) * DD + kt * 16 + col];
      v2f af[4];
#pragma unroll
      for (int k4 = 0; k4 < 4; ++k4) {
        af[k4].x = st[r * 17 + 4 * k4 + 2 * hf];
        af[k4].y = st[r * 17 + 4 * k4 + 2 * hf + 1];
      }
#pragma unroll
      for (int k4 = 0; k4 < 4; ++k4) {
        const int ks = kt * 4 + k4;
#pragma unroll
        for (int nn = 0; nn < 8; ++nn) {
          const v2f bf = *(const v2f*)(ldsW0 + ((ks * 8 + nn) << 6) + (lane << 1));
          acc0[nn] = wmma_f32_k4(af[k4], bf, acc0[nn]);
        }
      }
      // ---- acc1 += V0-tile(kt) @ ex1 ----
#pragma unroll
      for (int v = 0; v < 8; ++v)
        st[(v + 8 * hf) * 17 + col] =
            V0[rbase + (long)(v + 8 * hf) * DD + kt * 16 + col];
#pragma unroll
      for (int k4 = 0; k4 < 4; ++k4) {
        af[k4].x = st[r * 17 + 4 * k4 + 2 * hf];
        af[k4].y = st[r * 17 + 4 * k4 + 2 * hf + 1];
      }
#pragma unroll
      for (int k4 = 0; k4 < 4; ++k4) {
        const int ks = kt * 4 + k4;
#pragma unroll
        for (int nn = 0; nn < 8; ++nn) {
          const v2f bf = *(const v2f*)(ldsW1 + ((ks * 8 + nn) << 6) + (lane << 1));
          acc1[nn] = wmma_f32_k4(af[k4], bf, acc1[nn]);
        }
      }
    }

    // Gated combine, in-place overwrite (tile-local, so safe within the wave).
#pragma unroll
    for (int nn = 0; nn < 8; ++nn)
#pragma unroll
      for (int v = 0; v < 8; ++v) {
        const long idx = rbase + (long)(v + 8 * hf) * DD + nn * 16 + col;
        const float a0 = V0[idx];
        const float a1 = V1[idx];
        V0[idx] = a0 + t0 * silu_f(acc0[nn][v]);
        V1[idx] = a1 + t1 * silu_f(acc1[nn][v]);
      }
  }
}

// ---------------------------------------------------------------------------
// Kernel 4: edge MLP. E = silu(feat@e_w1+b1)@e_w2 + b2, feat=[d,-d,|d|] (7->pad 8).
// ---------------------------------------------------------------------------
__global__ void __launch_bounds__(256) edge_kernel(
    const float* __restrict__ node_pos, const long long* __restrict__ edges,
    const float* __restrict__ w1, const float* __restrict__ b1,
    const float* __restrict__ w2, const float* __restrict__ b2,
    float* __restrict__ outE) {
  __shared__ __align__(16) float ldsW2[DD * DD];
  __shared__ __align__(16) float stageAll[8][16 * 17];
  const int tid = threadIdx.x, wave = tid >> 5, lane = tid & 31;
  const int hf = lane >> 4, col = lane & 15;

  stage_weight_frags(ldsW2, w2, DD * DD, tid, 256);
  __syncthreads();

  // Layer-1 B fragments in registers, zero-padded beyond row 6 (K=7 -> 8).
  v2f w1f[2][8];
  float b1r[8], b2r[8];
#pragma unroll
  for (int s = 0; s < 2; ++s)
#pragma unroll
    for (int nn = 0; nn < 8; ++nn) {
      const int r0 = 4 * s + 2 * hf;
      w1f[s][nn].x = (r0 < 7) ? w1[r0 * DD + nn * 16 + col] : 0.0f;
      w1f[s][nn].y = (r0 + 1 < 7) ? w1[(r0 + 1) * DD + nn * 16 + col] : 0.0f;
    }
#pragma unroll
  for (int nn = 0; nn < 8; ++nn) {
    b1r[nn] = b1[nn * 16 + col];
    b2r[nn] = b2[nn * 16 + col];
  }

  const int tilesPerB = MEDGES / 16;
  const int nTiles = BB * tilesPerB;
  const int stride = gridDim.x * 8;
  float* st = stageAll[wave];

  for (int tile = blockIdx.x * 8 + wave; tile < nTiles; tile += stride) {
    const int b = tile / tilesPerB;
    const int e0 = (tile - b * tilesPerB) * 16;

    // Prefetch next grid-stride tile's edge indices (global_prefetch_b8).
    const int tnext = tile + stride;
    if (tnext < nTiles && lane < 16) {
      const int bn = tnext / tilesPerB;
      const long en = ((long)bn * MEDGES + (tnext - bn * tilesPerB) * 16 + lane) * 2;
      __builtin_prefetch(edges + en, 0, 3);
    }

    if (lane < 16) {
      const long ebase = ((long)b * MEDGES + e0 + lane) * 2;
      const long si = (long)edges[ebase];
      const long ri = (long)edges[ebase + 1];
      const float* ps = node_pos + ((long)b * NNODES + si) * 3;
      const float* pr = node_pos + ((long)b * NNODES + ri) * 3;
      const float d0 = pr[0] - ps[0];
      const float d1 = pr[1] - ps[1];
      const float d2 = pr[2] - ps[2];
      const float nrm = sqrtf(d0 * d0 + d1 * d1 + d2 * d2);
      float* sr = st + lane * 17;
      sr[0] = d0;  sr[1] = d1;  sr[2] = d2;
      sr[3] = -d0; sr[4] = -d1; sr[5] = -d2;
      sr[6] = nrm; sr[7] = 0.0f;
    }
    // Same-wave LDS ops are in-order; compiler inserts the dscnt wait.
    const int r = lane & 15;
    v2f a0, a1;
    a0.x = st[r * 17 + 2 * hf];     a0.y = st[r * 17 + 2 * hf + 1];
    a1.x = st[r * 17 + 4 + 2 * hf]; a1.y = st[r * 17 + 4 + 2 * hf + 1];

    v8f h[8], acc[8];
#pragma unroll
    for (int nn = 0; nn < 8; ++nn) {
      v8f c = {};
      c = wmma_f32_k4(a0, w1f[0][nn], c);
      c = wmma_f32_k4(a1, w1f[1][nn], c);
#pragma unroll
      for (int e = 0; e < 8; ++e) h[nn][e] = silu_f(c[e] + b1r[nn]);
#pragma unroll
      for (int e = 0; e < 8; ++e) acc[nn][e] = b2r[nn];
    }

    gemm_h_times_w(st, ldsW2, lane, h, acc);

    const long obase = ((long)b * MEDGES + e0) * DD;
#pragma unroll
    for (int nn = 0; nn < 8; ++nn)
#pragma unroll
      for (int v = 0; v < 8; ++v)
        outE[obase + (long)(v + 8 * hf) * DD + nn * 16 + col] = acc[nn][v];
  }
}

// ---------------------------------------------------------------------------
extern "C" void kernel_launch(void* const* d_in, const int* in_sizes, int n_in,
                              void* d_out, int out_size, void* d_ws, size_t ws_size,
                              hipStream_t stream) {
  (void)in_sizes; (void)n_in; (void)out_size; (void)ws_size;

  const float*     node_pos   = (const float*)d_in[0];
  const float*     state_in   = (const float*)d_in[1];
  const float*     time_i     = (const float*)d_in[2];
  const float*     conditions = (const float*)d_in[3];
  const long long* edges      = (const long long*)d_in[4];
  const float* f0_w1 = (const float*)d_in[5];
  const float* f0_b1 = (const float*)d_in[6];
  const float* f0_w2 = (const float*)d_in[7];
  const float* f0_b2 = (const float*)d_in[8];
  const float* f1_w1 = (const float*)d_in[9];
  const float* f1_b1 = (const float*)d_in[10];
  const float* f1_w2 = (const float*)d_in[11];
  const float* f1_b2 = (const float*)d_in[12];
  const float* t_w1  = (const float*)d_in[13];
  const float* t_b1  = (const float*)d_in[14];
  const float* t_w2  = (const float*)d_in[15];
  const float* t_b2  = (const float*)d_in[16];
  const float* c_w1  = (const float*)d_in[17];
  const float* c_b1  = (const float*)d_in[18];
  const float* c_w2  = (const float*)d_in[19];
  const float* c_b2  = (const float*)d_in[20];
  const float* e_w1  = (const float*)d_in[21];
  const float* e_b1  = (const float*)d_in[22];
  const float* e_w2  = (const float*)d_in[23];
  const float* e_b2  = (const float*)d_in[24];
  const float* ex0_w = (const float*)d_in[25];
  const float* ex0_b = (const float*)d_in[26];
  const float* ex1_w = (const float*)d_in[27];
  const float* ex1_b = (const float*)d_in[28];
  const float* gate0 = (const float*)d_in[29];
  const float* gate1 = (const float*)d_in[30];

  float* out = (float*)d_out;
  const long vlen = (long)BB * NNODES * DD;     // 12,800,000
  float* Vx0 = out;
  float* Vx1 = out + vlen;
  float* E   = out + 2 * vlen;
  float* biasTC = (float*)d_ws;                 // 2*128 floats of scratch

  bias_kernel<<<BB, 128, 0, stream>>>(time_i, conditions,
                                      t_w1, t_b1, t_w2, t_b2,
                                      c_w1, c_b1, c_w2, c_b2, biasTC);

  const int nodeTiles = BB * (NNODES / 16);     // 6250
  const int nodeWG = (nodeTiles + 7) / 8;       // 8 waves (tiles) per WG

  node_field_kernel<<<nodeWG, 256, 0, stream>>>(node_pos, state_in,
                                                f0_w1, f0_b1, f0_w2, f0_b2,
                                                biasTC, Vx0, 0);
  node_field_kernel<<<nodeWG, 256, 0, stream>>>(node_pos, state_in,
                                                f1_w1, f1_b1, f1_w2, f1_b2,
                                                biasTC, Vx1, 1);
  exchange_kernel<<<nodeWG, 256, 0, stream>>>(Vx0, Vx1, ex0_w, ex0_b,
                                              ex1_w, ex1_b, gate0, gate1);
  edge_kernel<<<2048, 256, 0, stream>>>(node_pos, edges,
                                        e_w1, e_b1, e_w2, e_b2, E);
}